// GNNPolicyNetworkWithMemory_18614388260923
// MI455X (gfx1250) — compile-verified
//
#include <hip/hip_runtime.h>
#include <hip/hip_bf16.h>
#include <math.h>

// ---------------------------------------------------------------------------
// GNN policy network w/ memory, gfx1250 (MI455X).
//   - edge aggregation: float4 gather + hardware global_atomic_add_f32 (L2-resident)
//   - GIN MLPs: fused 2x GEMM(64x64) per node tile via V_WMMA_F32_16X16X4_F32
//   - pooling: atomic mean-pool; LSTM gates: VALU; softmax: one wave32 per graph
// ---------------------------------------------------------------------------

typedef __attribute__((ext_vector_type(2))) float v2f;
typedef __attribute__((ext_vector_type(8))) float v8f;

#define NNODES 50000
#define NEDGES 800000
#define DF     64
#define BGRAPH 64
#define HLS    128
#define OUTC   32
#define WPAD   66   // 64 + 2 pad: breaks 64-bank stride conflicts, keeps 8B align

// ---------------- simple copy / zero ---------------------------------------
__global__ __launch_bounds__(256) void copy_f4(const float* __restrict__ src,
                                               float* __restrict__ dst, int n4) {
  int t = blockIdx.x * 256 + threadIdx.x;
  if (t < n4) ((float4*)dst)[t] = ((const float4*)src)[t];
}

__global__ __launch_bounds__(256) void zero_f(float* __restrict__ p, int n) {
  int t = blockIdx.x * 256 + threadIdx.x;
  if (t < n) p[t] = 0.0f;
}

// ---------------- edge scatter-add: agg[dst] += feat[src] ------------------
__global__ __launch_bounds__(256) void edge_scatter_add(
    const float* __restrict__ feat, const int* __restrict__ srcIdx,
    const int* __restrict__ dstIdx, float* __restrict__ accum, int n_edges) {
  int t = blockIdx.x * 256 + threadIdx.x;
  int e = t >> 4;                       // 16 threads per edge
  if (e >= n_edges) return;
  int c = (t & 15) << 2;                // 4 floats per thread
  int s = srcIdx[e];
  int d = dstIdx[e];
  const float4 v = *(const float4*)(feat + (size_t)s * DF + c);
  float* p = accum + (size_t)d * DF + c;
  unsafeAtomicAdd(p + 0, v.x);          // global_atomic_add_f32
  unsafeAtomicAdd(p + 1, v.y);
  unsafeAtomicAdd(p + 2, v.z);
  unsafeAtomicAdd(p + 3, v.w);
}

// ---------------- fused GIN MLP: out = relu(relu(hin@wA^T+bA)@wB^T+bB) -----
// 256 threads = 8 waves; each wave computes a 16-node x 64-feature tile.
// f32 WMMA operand mapping (wave32):
//   A 16x4 : lane m=lane&15; VGPR pair holds K = koff..koff+1, koff=(lane>=16)?2:0
//   B 4x16 : lane n=lane&15; same K pattern -> B[k][n]=W[n][k] is a float2 load
//            from row-major W at [n][k0+koff]
//   C/D    : vgpr i -> row i + ((lane>=16)?8:0), col = lane&15
__global__ __launch_bounds__(256) void gin_mlp_wmma(
    const float* __restrict__ hin, const float* __restrict__ wA,
    const float* __restrict__ bA, const float* __restrict__ wB,
    const float* __restrict__ bB, float* __restrict__ hout, int n_nodes) {
  __shared__ float sW1[DF * WPAD];
  __shared__ float sW2[DF * WPAD];
  __shared__ float sMid[8][16 * WPAD];   // per-wave intermediate tile (A layout)

  for (int idx = threadIdx.x; idx < DF * DF; idx += 256) {
    int n = idx >> 6, k = idx & 63;
    sW1[n * WPAD + k] = wA[idx];
    sW2[n * WPAD + k] = wB[idx];
  }
  __syncthreads();

  const int wv    = threadIdx.x >> 5;
  const int lane  = threadIdx.x & 31;
  const int m     = lane & 15;
  const int koff  = (lane >> 4) << 1;    // 0 or 2
  const int rbase = (lane >> 4) << 3;    // 0 or 8
  const int tile  = (blockIdx.x * 8 + wv) * 16;

  int nodeA = tile + m;
  if (nodeA >= n_nodes) nodeA = n_nodes - 1;   // clamp: keep EXEC all-1 for WMMA
  const float* arow = hin + (size_t)nodeA * DF + koff;

  // GEMM 1: mid = relu(hin @ wA^T + bA)
  v8f acc[4] = {};
#pragma unroll
  for (int k0 = 0; k0 < DF; k0 += 4) {
    v2f a = *(const v2f*)(arow + k0);
#pragma unroll
    for (int nt = 0; nt < 4; ++nt) {
      v2f b = *(const v2f*)(&sW1[(nt * 16 + m) * WPAD + koff + k0]);
      acc[nt] = __builtin_amdgcn_wmma_f32_16x16x4_f32(
          false, a, false, b, (short)0, acc[nt], false, false);
    }
  }

  float* mid = sMid[wv];
#pragma unroll
  for (int nt = 0; nt < 4; ++nt) {
    const int n = nt * 16 + m;
    const float bias = bA[n];
#pragma unroll
    for (int i = 0; i < 8; ++i) {
      float v = acc[nt][i] + bias;
      mid[(rbase + i) * WPAD + n] = v > 0.0f ? v : 0.0f;
    }
  }
  // same-wave LDS RAW: DS ops are in-order per wave; just stop compiler reorder
  asm volatile("" ::: "memory");

  // GEMM 2: out = relu(mid @ wB^T + bB)
  const float* a2 = &mid[m * WPAD + koff];
  v8f acc2[4] = {};
#pragma unroll
  for (int k0 = 0; k0 < DF; k0 += 4) {
    v2f a = *(const v2f*)(a2 + k0);
#pragma unroll
    for (int nt = 0; nt < 4; ++nt) {
      v2f b = *(const v2f*)(&sW2[(nt * 16 + m) * WPAD + koff + k0]);
      acc2[nt] = __builtin_amdgcn_wmma_f32_16x16x4_f32(
          false, a, false, b, (short)0, acc2[nt], false, false);
    }
  }

#pragma unroll
  for (int nt = 0; nt < 4; ++nt) {
    const int n = nt * 16 + m;
    const float bias = bB[n];
#pragma unroll
    for (int i = 0; i < 8; ++i) {
      int node = tile + rbase + i;
      if (node < n_nodes) {
        float v = acc2[nt][i] + bias;
        hout[(size_t)node * DF + n] = v > 0.0f ? v : 0.0f;  // outer ReLU
      }
    }
  }
}

// ---------------- mean pool over sorted batch ------------------------------
__global__ __launch_bounds__(256) void pool_scatter(
    const float* __restrict__ h, const int* __restrict__ batch,
    float* __restrict__ sums, float* __restrict__ cnt, int n_nodes) {
  int t = blockIdx.x * 256 + threadIdx.x;
  int node = t >> 4;
  if (node >= n_nodes) return;
  int c = (t & 15) << 2;
  int b = batch[node];
  const float4 v = *(const float4*)(h + (size_t)node * DF + c);
  float* p = sums + b * DF + c;
  unsafeAtomicAdd(p + 0, v.x);
  unsafeAtomicAdd(p + 1, v.y);
  unsafeAtomicAdd(p + 2, v.z);
  unsafeAtomicAdd(p + 3, v.w);
  if ((t & 15) == 0) unsafeAtomicAdd(cnt + b, 1.0f);
}

__global__ __launch_bounds__(256) void pool_div(const float* __restrict__ sums,
                                                const float* __restrict__ cnt,
                                                float* __restrict__ pooled) {
  int t = blockIdx.x * 256 + threadIdx.x;
  if (t >= BGRAPH * DF) return;
  float c = cnt[t >> 6];
  pooled[t] = sums[t] / fmaxf(c, 1.0f);
}

// ---------------- LSTM gates: pooled@Wih^T + h0@Whh^T + biases -------------
__global__ __launch_bounds__(256) void lstm_gates(
    const float* __restrict__ pooled, const float* __restrict__ h0,
    const float* __restrict__ wih, const float* __restrict__ whh,
    const float* __restrict__ bih, const float* __restrict__ bhh,
    float* __restrict__ gates) {
  int t = blockIdx.x * 256 + threadIdx.x;
  if (t >= BGRAPH * 4 * HLS) return;
  int b = t >> 9;
  int j = t & 511;
  float acc = bih[j] + bhh[j];
  const float* pr = pooled + b * DF;
  const float* wr = wih + j * DF;
#pragma unroll 8
  for (int k = 0; k < DF; ++k) acc += pr[k] * wr[k];
  const float* hr = h0 + b * HLS;
  const float* wh = whh + j * HLS;
#pragma unroll 8
  for (int k = 0; k < HLS; ++k) acc += hr[k] * wh[k];
  gates[t] = acc;
}

__device__ __forceinline__ float sigf(float x) { return 1.0f / (1.0f + __expf(-x)); }

__global__ __launch_bounds__(256) void lstm_cell(
    const float* __restrict__ gates, const float* __restrict__ c0,
    float* __restrict__ h1, float* __restrict__ c1) {
  int t = blockIdx.x * 256 + threadIdx.x;
  if (t >= BGRAPH * HLS) return;
  int b = t >> 7, h = t & 127;
  const float* g = gates + b * 4 * HLS;          // PyTorch gate order: i,f,g,o
  float ig = sigf(g[h]);
  float fg = sigf(g[HLS + h]);
  float gg = tanhf(g[2 * HLS + h]);
  float og = sigf(g[3 * HLS + h]);
  float c = fg * c0[t] + ig * gg;
  c1[t] = c;
  h1[t] = og * tanhf(c);
}

// ---------------- FC + softmax: one wave32 per graph (OUT == 32) -----------
__global__ __launch_bounds__(32) void fc_softmax(
    const float* __restrict__ h1, const float* __restrict__ fcw,
    const float* __restrict__ fcb, float* __restrict__ probs) {
  int b = blockIdx.x;
  int lane = threadIdx.x;
  float acc = fcb[lane];
  const float* hr = h1 + b * HLS;
  const float* wr = fcw + lane * HLS;
#pragma unroll 8
  for (int k = 0; k < HLS; ++k) acc += hr[k] * wr[k];
  float mx = acc;
#pragma unroll
  for (int off = 16; off; off >>= 1) mx = fmaxf(mx, __shfl_xor(mx, off, 32));
  float e = __expf(acc - mx);
  float s = e;
#pragma unroll
  for (int off = 16; off; off >>= 1) s += __shfl_xor(s, off, 32);
  probs[b * OUTC + lane] = e / s;
}

// ---------------------------------------------------------------------------
extern "C" void kernel_launch(void* const* d_in, const int* in_sizes, int n_in,
                              void* d_out, int out_size, void* d_ws, size_t ws_size,
                              hipStream_t stream) {
  const float* x    = (const float*)d_in[0];
  const int*   ei   = (const int*)d_in[1];
  const int*   bat  = (const int*)d_in[2];
  const float *w1 = (const float*)d_in[3],  *b1 = (const float*)d_in[4];
  const float *w2 = (const float*)d_in[5],  *b2 = (const float*)d_in[6];
  const float *w3 = (const float*)d_in[7],  *b3 = (const float*)d_in[8];
  const float *w4 = (const float*)d_in[9],  *b4 = (const float*)d_in[10];
  const float *wih = (const float*)d_in[11], *whh = (const float*)d_in[12];
  const float *bih = (const float*)d_in[13], *bhh = (const float*)d_in[14];
  const float *fcw = (const float*)d_in[15], *fcb = (const float*)d_in[16];
  const float *h0 = (const float*)d_in[17],  *c0 = (const float*)d_in[18];

  // workspace: 2 node buffers (12.8 MB each) + pooling/LSTM tails (~26 MB total)
  float* ws     = (float*)d_ws;
  float* bufA   = ws;
  float* bufB   = bufA + (size_t)NNODES * DF;
  float* sums   = bufB + (size_t)NNODES * DF;
  float* cnt    = sums + BGRAPH * DF;         // contiguous after sums
  float* pooled = cnt + BGRAPH;
  float* gates  = pooled + BGRAPH * DF;

  // d_out layout: probs[64,32] | h1[1,64,128] | c1[1,64,128]
  float* probs = (float*)d_out;
  float* h1out = probs + BGRAPH * OUTC;
  float* c1out = h1out + BGRAPH * HLS;

  const int* srcI = ei;
  const int* dstI = ei + NEDGES;

  const int nCopy4 = NNODES * DF / 4;
  const int copyB  = (nCopy4 + 255) / 256;
  const int edgeB  = (NEDGES * 16 + 255) / 256;
  const int mlpB   = (NNODES + 127) / 128;    // 128 nodes per 8-wave block

  // ---- GIN layer 1 ----
  copy_f4<<<copyB, 256, 0, stream>>>(x, bufA, nCopy4);
  edge_scatter_add<<<edgeB, 256, 0, stream>>>(x, srcI, dstI, bufA, NEDGES);
  gin_mlp_wmma<<<mlpB, 256, 0, stream>>>(bufA, w1, b1, w2, b2, bufB, NNODES);

  // ---- GIN layer 2 ----
  copy_f4<<<copyB, 256, 0, stream>>>(bufB, bufA, nCopy4);
  edge_scatter_add<<<edgeB, 256, 0, stream>>>(bufB, srcI, dstI, bufA, NEDGES);
  gin_mlp_wmma<<<mlpB, 256, 0, stream>>>(bufA, w3, b3, w4, b4, bufB, NNODES);

  // ---- global mean pool ----
  zero_f<<<(BGRAPH * DF + BGRAPH + 255) / 256, 256, 0, stream>>>(sums, BGRAPH * DF + BGRAPH);
  pool_scatter<<<(NNODES * 16 + 255) / 256, 256, 0, stream>>>(bufB, bat, sums, cnt, NNODES);
  pool_div<<<(BGRAPH * DF + 255) / 256, 256, 0, stream>>>(sums, cnt, pooled);

  // ---- LSTM step + head ----
  lstm_gates<<<(BGRAPH * 4 * HLS + 255) / 256, 256, 0, stream>>>(
      pooled, h0, wih, whh, bih, bhh, gates);
  lstm_cell<<<(BGRAPH * HLS + 255) / 256, 256, 0, stream>>>(gates, c0, h1out, c1out);
  fc_softmax<<<BGRAPH, 32, 0, stream>>>(h1out, fcw, fcb, probs);
}